// GCNMultiLabel_84567906058704
// MI455X (gfx1250) — compile-verified
//
#include <hip/hip_runtime.h>
#include <math.h>

typedef __attribute__((ext_vector_type(2))) float v2f;
typedef __attribute__((ext_vector_type(8))) float v8f;

#define N_NODES 100000
#define N_EDGES 1600000
#define IN_F 256
#define H_F 128
#define N_CLS 64

// ---------------------------------------------------------------------------
// Degree accumulation: deg_src (out-degree), deg_dst (in-degree)
// ---------------------------------------------------------------------------
__global__ void deg_kernel(const int* __restrict__ src, const int* __restrict__ dst,
                           float* __restrict__ deg_src, float* __restrict__ deg_dst,
                           int E) {
    int i = blockIdx.x * blockDim.x + threadIdx.x;
    if (i < E) {
        atomicAdd(&deg_src[src[i]], 1.0f);
        atomicAdd(&deg_dst[dst[i]], 1.0f);
    }
}

// In-place: deg -> (deg > 0 ? rsqrt(deg) : 0)
__global__ void norm_kernel(float* __restrict__ d, int n) {
    int i = blockIdx.x * blockDim.x + threadIdx.x;
    if (i < n) {
        float v = d[i];
        d[i] = (v > 0.0f) ? rsqrtf(v) : 0.0f;
    }
}

// ---------------------------------------------------------------------------
// WMMA fp32 GEMM with M-register-blocking:
//   out[m][n] = (A[m][:] . W[:][n]) * scale[m]
//
// One wave owns MT row-tiles (MT*16 rows) x one 16-col tile. Per K-step of 4
// it loads ONE B fragment and MT A fragments, issuing MT back-to-back WMMAs
// -> B traffic cut by MT, wmma density in the instruction stream up ~3x.
// Block has N/16 waves covering all N columns of the row strip.
// M must be a multiple of MT*16 (100000 = 1250*80), so EXEC is all-1s at
// every v_wmma (ISA requirement).
//
// Fragment layouts per CDNA5 ISA 7.12.2 (wave32):
//   A 16x4 f32 : lanes 0-15 hold M=lane, K={0,1}; lanes 16-31 hold K={2,3}
//   B 4x16 f32 : lanes 0-15 hold N=lane, K={0,1}; lanes 16-31 hold K={2,3}
//   C/D 16x16  : 8 VGPRs; lane<16 -> N=lane, M=r; lane>=16 -> N=lane-16, M=8+r
// ---------------------------------------------------------------------------
template<int K, int N, int MT>
__global__ void gemm_wmma_scale_kernel(const float* __restrict__ A,     // [M][K]
                                       const float* __restrict__ W,     // [K][N]
                                       const float* __restrict__ scale, // [M]
                                       float* __restrict__ out) {       // [M][N]
    const int wave  = threadIdx.x >> 5;
    const int lane  = threadIdx.x & 31;
    const int lhalf = lane >> 4;       // 0: K pair {0,1}; 1: K pair {2,3}
    const int l16   = lane & 15;
    const int row0  = blockIdx.x * (16 * MT);
    const int col0  = wave * 16;

    const int koff = lhalf ? 2 : 0;
    const float* arow = A + (size_t)(row0 + l16) * K + koff;     // A[row][k+koff]
    const float* wcol = W + (size_t)koff * N + (col0 + l16);     // W[k+koff][col]

    v8f c[MT];
#pragma unroll
    for (int t = 0; t < MT; ++t) c[t] = (v8f){};

    for (int k = 0; k < K; k += 4) {
        v2f b;
        b.x = wcol[(size_t)k * N];
        b.y = wcol[(size_t)(k + 1) * N];
#pragma unroll
        for (int t = 0; t < MT; ++t) {
            v2f a;
            // row-tile offset t*16*K folds into the 24-bit immediate offset
            a.x = arow[k + (size_t)t * 16 * K];
            a.y = arow[k + 1 + (size_t)t * 16 * K];
            // D = A x B + C   (fp32, 16x16x4)
            c[t] = __builtin_amdgcn_wmma_f32_16x16x4_f32(
                    /*neg_a=*/false, a, /*neg_b=*/false, b,
                    /*c_mod=*/(short)0, c[t], /*reuse_a=*/false, /*reuse_b=*/false);
        }
    }

#pragma unroll
    for (int t = 0; t < MT; ++t) {
#pragma unroll
        for (int r = 0; r < 8; ++r) {
            int m = row0 + t * 16 + r + (lhalf ? 8 : 0);
            out[(size_t)m * N + col0 + l16] = c[t][r] * scale[m];
        }
    }
}

// ---------------------------------------------------------------------------
// Edge scatter-add: agg[dst[e]][:] += h[src[e]][:]   (F floats per edge)
// F/4 float4 chunks per edge; consecutive threads handle consecutive chunks of
// one edge -> coalesced 128B gathers; atomics land in the 192MB L2 (h and agg
// both fit on-chip).
// ---------------------------------------------------------------------------
template<int F>
__global__ void edge_agg_kernel(const int* __restrict__ src, const int* __restrict__ dst,
                                const float* __restrict__ h, float* __restrict__ agg,
                                int E) {
    constexpr int CH = F / 4;
    long long tid = (long long)blockIdx.x * blockDim.x + threadIdx.x;
    if (tid >= (long long)E * CH) return;
    int e = (int)(tid / CH);
    int j = (int)(tid % CH) * 4;
    int s = src[e];
    int d = dst[e];
    const float4 v = *(const float4*)(h + (size_t)s * F + j);
    float* o = agg + (size_t)d * F + j;
    atomicAdd(o + 0, v.x);
    atomicAdd(o + 1, v.y);
    atomicAdd(o + 2, v.z);
    atomicAdd(o + 3, v.w);
}

// out = relu(agg * norm_dst[row] + b[col])
template<int F>
__global__ void bias_relu_kernel(const float* __restrict__ agg,
                                 const float* __restrict__ norm_dst,
                                 const float* __restrict__ b,
                                 float* __restrict__ out, int M) {
    long long i = (long long)blockIdx.x * blockDim.x + threadIdx.x;
    if (i < (long long)M * F) {
        int row = (int)(i / F);
        int col = (int)(i % F);
        float v = agg[i] * norm_dst[row] + b[col];
        out[i] = fmaxf(v, 0.0f);
    }
}

// out = sigmoid(agg * norm_dst[row] + b[col])
template<int F>
__global__ void bias_sigmoid_kernel(const float* __restrict__ agg,
                                    const float* __restrict__ norm_dst,
                                    const float* __restrict__ b,
                                    float* __restrict__ out, int M) {
    long long i = (long long)blockIdx.x * blockDim.x + threadIdx.x;
    if (i < (long long)M * F) {
        int row = (int)(i / F);
        int col = (int)(i % F);
        float v = agg[i] * norm_dst[row] + b[col];
        out[i] = 1.0f / (1.0f + __expf(-v));
    }
}

// ---------------------------------------------------------------------------
// Orchestration. Workspace layout (floats):
//   [0, N)          deg_src  -> norm_src after normalize
//   [N, 2N)         deg_dst  -> norm_dst
//   h_buf  = 2N     .. 2N + N*128    : h_scaled, then h1(relu), then agg2 (N*64)
//   a_buf  = h_buf + N*128 .. +N*128 : agg1, then t_scaled (N*64)
// Peak usage: (2*N_NODES + 2*N_NODES*128)*4 bytes ~= 103.2 MB
// ---------------------------------------------------------------------------
extern "C" void kernel_launch(void* const* d_in, const int* in_sizes, int n_in,
                              void* d_out, int out_size, void* d_ws, size_t ws_size,
                              hipStream_t stream) {
    const float* feat = (const float*)d_in[0];
    const int*   src  = (const int*)d_in[1];
    const int*   dst  = (const int*)d_in[2];
    const float* W1   = (const float*)d_in[3];
    const float* b1   = (const float*)d_in[4];
    const float* W2   = (const float*)d_in[5];
    const float* b2   = (const float*)d_in[6];
    float* out = (float*)d_out;

    float* ws       = (float*)d_ws;
    float* deg_src  = ws;                       // becomes norm_src
    float* deg_dst  = ws + N_NODES;             // becomes norm_dst
    float* h_buf    = ws + 2 * (size_t)N_NODES;                 // N*128
    float* a_buf    = h_buf + (size_t)N_NODES * H_F;            // N*128
    float* agg2     = h_buf;                    // reuse (N*64), after h1 consumed
    float* t_buf    = a_buf;                    // reuse (N*64), after agg1 consumed

    const int TPB = 256;
    constexpr int MT = 5;                       // 100000 = 1250 * (16*5)

    // Zero degree counters and layer-1 aggregation buffer (must happen every
    // call: graph replay does not re-poison).
    hipMemsetAsync(deg_src, 0, 2 * (size_t)N_NODES * sizeof(float), stream);
    hipMemsetAsync(a_buf, 0, (size_t)N_NODES * H_F * sizeof(float), stream);

    // Degrees + rsqrt norms
    deg_kernel<<<(N_EDGES + TPB - 1) / TPB, TPB, 0, stream>>>(src, dst, deg_src, deg_dst, N_EDGES);
    norm_kernel<<<(2 * N_NODES + TPB - 1) / TPB, TPB, 0, stream>>>(deg_src, 2 * N_NODES);

    // Layer 1: h_scaled = (feat @ W1) * norm_src   (WMMA fp32, 80-row blocks)
    gemm_wmma_scale_kernel<IN_F, H_F, MT>
        <<<N_NODES / (16 * MT), (H_F / 16) * 32, 0, stream>>>(feat, W1, deg_src, h_buf);

    // agg1[dst] += h_scaled[src]
    {
        long long total = (long long)N_EDGES * (H_F / 4);
        edge_agg_kernel<H_F><<<(int)((total + TPB - 1) / TPB), TPB, 0, stream>>>(
            src, dst, h_buf, a_buf, N_EDGES);
    }

    // h1 = relu(agg1 * norm_dst + b1)
    {
        long long total = (long long)N_NODES * H_F;
        bias_relu_kernel<H_F><<<(int)((total + TPB - 1) / TPB), TPB, 0, stream>>>(
            a_buf, deg_dst, b1, h_buf, N_NODES);
    }

    // Layer 2 GEMM: t_scaled = (h1 @ W2) * norm_src  (WMMA fp32)
    gemm_wmma_scale_kernel<H_F, N_CLS, MT>
        <<<N_NODES / (16 * MT), (N_CLS / 16) * 32, 0, stream>>>(h_buf, W2, deg_src, t_buf);

    // agg2[dst] += t_scaled[src]   (agg2 reuses h_buf region, zero it first)
    hipMemsetAsync(agg2, 0, (size_t)N_NODES * N_CLS * sizeof(float), stream);
    {
        long long total = (long long)N_EDGES * (N_CLS / 4);
        edge_agg_kernel<N_CLS><<<(int)((total + TPB - 1) / TPB), TPB, 0, stream>>>(
            src, dst, t_buf, agg2, N_EDGES);
    }

    // out = sigmoid(agg2 * norm_dst + b2)
    {
        long long total = (long long)N_NODES * N_CLS;
        bias_sigmoid_kernel<N_CLS><<<(int)((total + TPB - 1) / TPB), TPB, 0, stream>>>(
            agg2, deg_dst, b2, out, N_NODES);
    }
}